// CCHLoss_2602750181384
// MI455X (gfx1250) — compile-verified
//
#include <hip/hip_runtime.h>

// Reference dims
#define BATCH 8      // B_OUTER * N_VIEWS
#define PTS   4096   // H * W
#define DWCH  24

typedef __attribute__((ext_vector_type(2))) float v2f;
typedef __attribute__((ext_vector_type(8))) float v8f;

// ws float layout: [0]=sum(masked cham_x + cham_y), [1]=sum (vc-vc_pred)^2,
//                  [2]=sum mask, [3]=sum pred_dw^2

// ---------------------------------------------------------------------------
// Chamfer kernel: grid = (PTS/128 row-tiles, BATCH, 2 directions).
// dir 0: X=v_pred, Y=v  (cham_x, masked);  dir 1: X=v, Y=v_pred (cham_y).
//
// Trick: the K=4 pad slot of V_WMMA_F32_16X16X4_F32 carries the |y|^2 term:
//   A = [-2x0, -2x1, -2x2, 1.0]   (rows, pre-scaled)
//   B = [ y0,   y1,   y2, |y|^2]  (columns)
//   C = |x|^2 (constant per wave)
// => D[p][q] = |x_p|^2 - 2 x_p.y_q + |y_q|^2   in a single WMMA per tile.
//
// LDS stores Y as two float2 arrays so each lane-half fetches its K-pair with
// one ds_load_b64:  pairLo[q]=(y0,y1)  for lanes 0-15 (K0,K1),
//                   pairHi[q]=(y2,|y|^2) for lanes 16-31 (K2,K3).
// ---------------------------------------------------------------------------
__global__ __launch_bounds__(256) void chamfer_wmma_kernel(
    const float* __restrict__ v, const float* __restrict__ v_pred,
    const float* __restrict__ mask, float* __restrict__ ws)
{
    extern __shared__ float2 ypair[];       // 2*PTS float2 (64 KB)
    float2* pairLo = ypair;                 // (y0, y1)
    float2* pairHi = ypair + PTS;           // (y2, |y|^2)
    __shared__ float blockSum;

    const int b   = blockIdx.y;
    const int dir = blockIdx.z;
    const float* X = (dir == 0) ? v_pred : v;
    const float* Y = (dir == 0) ? v : v_pred;

    const int tid = threadIdx.x;
    if (tid == 0) blockSum = 0.0f;

    // Stage Y into LDS as K-pairs.
    const float* Yb = Y + (size_t)b * PTS * 3;
    for (int q = tid; q < PTS; q += 256) {
        float y0 = Yb[q * 3 + 0];
        float y1 = Yb[q * 3 + 1];
        float y2 = Yb[q * 3 + 2];
        pairLo[q] = make_float2(y0, y1);
        pairHi[q] = make_float2(y2, y0 * y0 + y1 * y1 + y2 * y2);
    }
    __syncthreads();

    const int lane    = tid & 31;
    const int m       = lane & 15;   // row / column index within 16-lane half
    const int khalf   = lane >> 4;   // 0 -> K{0,1}, 1 -> K{2,3}
    const int wave    = tid >> 5;
    const int rowbase = blockIdx.x * 128 + wave * 16;
    const int row     = rowbase + m;

    const float* Xb = X + (size_t)b * PTS * 3;
    const float x0  = Xb[row * 3 + 0];
    const float x1  = Xb[row * 3 + 1];
    const float x2c = Xb[row * 3 + 2];
    const float myx2 = x0 * x0 + x1 * x1 + x2c * x2c;

    // A (16x4): VGPR0 = K(0|2), VGPR1 = K(1|3); K3 slot carries the 1.0
    // that multiplies |y|^2.
    v2f A;
    A.x = -2.0f * (khalf ? x2c : x0);
    A.y = khalf ? 1.0f : (-2.0f * x1);

    // Constant C: |x|^2 of the 16 rows this wave owns.
    // C/D reg r holds rows {r | r+8} across the two lane halves.
    v8f C;
#pragma unroll
    for (int r = 0; r < 8; ++r)
        C[r] = __shfl(myx2, r + khalf * 8, 32);

    // Per-lane B base: one ds_load_b64 per tile from base + q0*8 bytes.
    const float2* __restrict__ bbase = (khalf ? pairHi : pairLo) + m;

    v8f acc;
#pragma unroll
    for (int r = 0; r < 8; ++r) acc[r] = 3.402823466e38f;

#pragma unroll 4
    for (int q0 = 0; q0 < PTS; q0 += 16) {
        const float2 bp = bbase[q0];
        v2f B;
        B.x = bp.x;
        B.y = bp.y;

        v8f D = __builtin_amdgcn_wmma_f32_16x16x4_f32(
            false, A, false, B, (short)0, C, false, false);

#pragma unroll
        for (int r = 0; r < 8; ++r) acc[r] = fminf(acc[r], D[r]);
    }

    // Min across the 16 columns held by each 16-lane group.
#pragma unroll
    for (int r = 0; r < 8; ++r) {
        float a = acc[r];
        a = fminf(a, __shfl_xor(a, 1, 32));
        a = fminf(a, __shfl_xor(a, 2, 32));
        a = fminf(a, __shfl_xor(a, 4, 32));
        a = fminf(a, __shfl_xor(a, 8, 32));
        acc[r] = a;
    }

    // lanes with m==0 (lane 0 and lane 16) hold row mins for rows
    // rowbase + r + khalf*8.
    if (m == 0) {
        float contrib = 0.0f;
#pragma unroll
        for (int r = 0; r < 8; ++r) {
            const int rr = rowbase + r + khalf * 8;
            const float w = (dir == 0) ? mask[b * PTS + rr] : 1.0f;
            contrib += acc[r] * w;
        }
        atomicAdd(&blockSum, contrib);   // ds_add_f32
    }
    __syncthreads();
    if (tid == 0) atomicAdd(&ws[0], blockSum);
}

// ---------------------------------------------------------------------------
// Elementwise reductions: sum (vc-vc_pred)^2, sum mask, sum pred_dw^2.
// ---------------------------------------------------------------------------
__global__ __launch_bounds__(256) void aux_sums_kernel(
    const float* __restrict__ vc, const float* __restrict__ vc_pred,
    const float* __restrict__ mask, const float* __restrict__ dw,
    float* __restrict__ ws)
{
    const int tid = blockIdx.x * blockDim.x + threadIdx.x;
    const int stride = gridDim.x * blockDim.x;
    float s_rgb = 0.0f, s_mask = 0.0f, s_dw = 0.0f;

    for (int i = tid; i < BATCH * PTS * 3; i += stride) {
        float d = vc[i] - vc_pred[i];
        s_rgb += d * d;
    }
    for (int i = tid; i < BATCH * PTS; i += stride)
        s_mask += mask[i];
    for (int i = tid; i < BATCH * PTS * DWCH; i += stride) {
        float d = dw[i];
        s_dw += d * d;
    }

#pragma unroll
    for (int off = 1; off < 32; off <<= 1) {
        s_rgb  += __shfl_xor(s_rgb,  off, 32);
        s_mask += __shfl_xor(s_mask, off, 32);
        s_dw   += __shfl_xor(s_dw,   off, 32);
    }

    __shared__ float red[3][8];
    const int lane = threadIdx.x & 31;
    const int wave = threadIdx.x >> 5;
    if (lane == 0) { red[0][wave] = s_rgb; red[1][wave] = s_mask; red[2][wave] = s_dw; }
    __syncthreads();
    if (threadIdx.x == 0) {
        float a = 0.0f, bb = 0.0f, c = 0.0f;
        for (int w = 0; w < 8; ++w) { a += red[0][w]; bb += red[1][w]; c += red[2][w]; }
        atomicAdd(&ws[1], a);
        atomicAdd(&ws[2], bb);
        atomicAdd(&ws[3], c);
    }
}

// ---------------------------------------------------------------------------
// Finalize: total, posed, canonical, loss_w  (canonical = mse * mean(mask)).
// ---------------------------------------------------------------------------
__global__ void finalize_kernel(const float* __restrict__ ws,
                                float* __restrict__ out)
{
    if (threadIdx.x == 0 && blockIdx.x == 0) {
        const float inv_bp = 1.0f / (float)(BATCH * PTS);
        const float posed     = ws[0] * inv_bp;
        const float mse       = ws[1] / (float)(BATCH * PTS * 3);
        const float meanmask  = ws[2] * inv_bp;
        const float canonical = mse * meanmask;
        const float lw        = ws[3] / (float)(BATCH * PTS * DWCH);
        out[0] = posed + canonical + lw;
        out[1] = posed;
        out[2] = canonical;
        out[3] = lw;
    }
}

extern "C" void kernel_launch(void* const* d_in, const int* in_sizes, int n_in,
                              void* d_out, int out_size, void* d_ws, size_t ws_size,
                              hipStream_t stream) {
    const float* v       = (const float*)d_in[0];
    const float* v_pred  = (const float*)d_in[1];
    const float* vc      = (const float*)d_in[2];
    const float* vc_pred = (const float*)d_in[3];
    const float* mask    = (const float*)d_in[4];
    const float* dw      = (const float*)d_in[5];
    float* ws  = (float*)d_ws;
    float* out = (float*)d_out;

    hipMemsetAsync(ws, 0, 4 * sizeof(float), stream);

    dim3 grid(PTS / 128, BATCH, 2);
    const size_t shmem = 2 * PTS * sizeof(float2);   // 64 KB dynamic LDS
    chamfer_wmma_kernel<<<grid, 256, shmem, stream>>>(v, v_pred, mask, ws);

    aux_sums_kernel<<<64, 256, 0, stream>>>(vc, vc_pred, mask, dw, ws);
    finalize_kernel<<<1, 32, 0, stream>>>(ws, out);
}